// EmbeddingModel_19353122635908
// MI455X (gfx1250) — compile-verified
//
#include <hip/hip_runtime.h>

typedef __attribute__((ext_vector_type(2))) float v2f;
typedef __attribute__((ext_vector_type(8))) float v8f;
typedef __attribute__((ext_vector_type(2))) int   v2i;

// Problem constants (match the reference)
#define V_SZ 100000
#define E_SZ 128
#define B_SZ 16384
#define C_SZ 10
#define K_SZ 64
#define S_SZ   (C_SZ + K_SZ)   // 74 slots per batch row
#define NPAIR  (S_SZ / 2)      // 37 aligned slot-pairs (C and K both even)
#define NWAVES 4               // waves per tile (per workgroup)

// One workgroup (4 wave32) handles a tile of 16 batch rows.
// A (16x128 f32) = input embeddings of the tile rows, preloaded per wave.
// Slot-pairs are strided across the 4 waves; each pair runs two independent
// chains of 32 x v_wmma_f32_16x16x4_f32 over E=128. Diagonals D[j][j] are the
// logits; per-wave partial losses combine via ds_add_f32 in LDS.
__global__ __launch_bounds__(128)
void w2v_nsloss_wmma(const int* __restrict__ inout_labels,
                     const int* __restrict__ near_labels,
                     const int* __restrict__ neg_labels,
                     const float* __restrict__ in_embed,
                     const float* __restrict__ out_embed,
                     float* __restrict__ out)
{
    __shared__ float tile_loss[16];

    const int lane = threadIdx.x & 31;
    const int wave = threadIdx.x >> 5;              // 0..3
    const int row_base = blockIdx.x * 16;

    const int col = lane & 15;   // matrix row/column index this lane serves
    const int hi  = lane >> 4;   // 0 -> K pair {0,1}, 1 -> K pair {2,3}
    const int b   = row_base + col;

    if (threadIdx.x < 16) tile_loss[threadIdx.x] = 0.0f;
    __syncthreads();

    // ---- Preload A fragments: lane holds in_row[col][4k + 2*hi + {0,1}] ----
    const int in_lab = inout_labels[row_base + col];
    const float* arow = in_embed + (size_t)in_lab * E_SZ + 2 * hi;
    v2f A[32];
#pragma unroll
    for (int kk = 0; kk < 32; ++kk)
        A[kk] = *(const v2f*)(arow + 4 * kk);

    // Row-loss accumulator: lane r (r<8) owns row r, lane r+24 owns row r+8.
    float rl = 0.0f;

    // Label fetch for a slot-pair: pair p covers slots {2p, 2p+1}.
    // p < 5  -> near pair (near_labels[b*10 + 2p]), p >= 5 -> neg pair.
    auto load_pair = [&](int p) -> v2i {
        const int* lp = (p >= C_SZ / 2)
                            ? (neg_labels  + (size_t)b * K_SZ + (2 * p - C_SZ))
                            : (near_labels + (size_t)b * C_SZ + 2 * p);
        return *(const v2i*)lp;   // 8B-aligned: even int32 index
    };

    int  p     = wave;                       // wave-strided pair index
    v2i  labs  = load_pair(p);
    bool isneg = (p >= C_SZ / 2);

    while (p < NPAIR) {
        // ---- software-pipeline: prefetch next pair's labels ----
        const int pn = p + NWAVES;
        v2i  labs_n  = labs;
        bool isneg_n = isneg;
        if (pn < NPAIR) {
            labs_n  = load_pair(pn);
            isneg_n = (pn >= C_SZ / 2);
        }

        // ---- two independent WMMA chains for slots 2p and 2p+1 ----
        const float* brow0 = out_embed + (size_t)labs.x * E_SZ + 2 * hi;
        const float* brow1 = out_embed + (size_t)labs.y * E_SZ + 2 * hi;
        v8f acc0 = {}, acc1 = {};
#pragma unroll
        for (int kk = 0; kk < 32; ++kk) {
            v2f b0 = *(const v2f*)(brow0 + 4 * kk);   // global_load_b64
            v2f b1 = *(const v2f*)(brow1 + 4 * kk);
            acc0 = __builtin_amdgcn_wmma_f32_16x16x4_f32(
                       false, A[kk], false, b0, (short)0, acc0, false, false);
            acc1 = __builtin_amdgcn_wmma_f32_16x16x4_f32(
                       false, A[kk], false, b1, (short)0, acc1, false, false);
        }

        // ---- diagonal extraction + logsigmoid (branch-free selects) ----
        const float sgn = isneg ? -1.0f : 1.0f;
#pragma unroll
        for (int r = 0; r < 8; ++r) {
            const bool diag = (lane == r) || (lane == r + 24);
            const float x0 = sgn * acc0[r];
            const float x1 = sgn * acc1[r];
            // stable log_sigmoid(x) = min(x,0) - log(1 + exp(-|x|))
            const float ls0 = fminf(x0, 0.0f) - __logf(1.0f + __expf(-fabsf(x0)));
            const float ls1 = fminf(x1, 0.0f) - __logf(1.0f + __expf(-fabsf(x1)));
            rl += diag ? (ls0 + ls1) : 0.0f;
        }

        labs  = labs_n;
        isneg = isneg_n;
        p     = pn;
    }

    // Fold lanes 24..31 (rows 8..15) down to lanes 8..15; 16 lanes per wave.
    const float other  = __shfl(rl, lane + 16, 32);
    const float result = (lane < 8) ? rl : other;

    // Cross-wave combine in LDS (ds_add_f32).
    if (lane < 16)
        atomicAdd(&tile_loss[lane], result);
    __syncthreads();

    if (threadIdx.x < 16)
        out[row_base + threadIdx.x] = -tile_loss[threadIdx.x];
}

extern "C" void kernel_launch(void* const* d_in, const int* in_sizes, int n_in,
                              void* d_out, int out_size, void* d_ws, size_t ws_size,
                              hipStream_t stream)
{
    (void)in_sizes; (void)n_in; (void)d_ws; (void)ws_size; (void)out_size;
    const int*   inout_labels = (const int*)  d_in[0];
    const int*   near_labels  = (const int*)  d_in[1];
    const int*   neg_labels   = (const int*)  d_in[2];
    const float* in_embed     = (const float*)d_in[3];
    const float* out_embed    = (const float*)d_in[4];
    float*       out          = (float*)d_out;

    // One 16-row tile per workgroup; 4 wave32 per workgroup -> 4096 waves.
    dim3 grid(B_SZ / 16);
    dim3 block(32 * NWAVES);
    w2v_nsloss_wmma<<<grid, block, 0, stream>>>(inout_labels, near_labels,
                                                neg_labels, in_embed,
                                                out_embed, out);
}